// Resample2d_67886253081023
// MI455X (gfx1250) — compile-verified
//
#include <hip/hip_runtime.h>

// Resample2d: out[b,c,i,j] = bilinear(img[b,c]) at flow-warped coords, border clamp.
// Memory-bound (~0.55 GB -> ~24us @ 23.3 TB/s HBM). Two adjacent pixels per
// thread: b64 NT flow loads + b64 NT output stores; all coordinate math and
// bilinear weights computed once per pixel and reused across the 32 channels.

typedef float f2 __attribute__((ext_vector_type(2)));

constexpr int B = 8, C = 32, H = 512, W = 512;
constexpr int HW = H * W;

__global__ __launch_bounds__(256) void resample2d_kernel(
    const float* __restrict__ img,    // (B,C,H,W)
    const float* __restrict__ flow,   // (B,2,H,W)
    float* __restrict__ out)          // (B,C,H,W)
{
    int gid = blockIdx.x * blockDim.x + threadIdx.x;   // one thread per (b,i,j-pair)
    if (gid >= (B * HW) / 2) return;

    const int j2 = (gid & 255) << 1;       // even j: 8B-aligned pair, coalesced lanes
    const int i  = (gid >> 8) & (H - 1);
    const int b  = gid >> 17;

    // Flow: read exactly once -> non-temporal b64 loads (TH_LOAD_NT).
    const float* fb = flow + (size_t)b * 2 * HW + (size_t)i * W + j2;
    const f2 fx = __builtin_nontemporal_load((const f2*)fb);
    const f2 fy = __builtin_nontemporal_load((const f2*)(fb + HW));

    const float inv   = 2.0f / (float)(W - 1);   // H == W == 512
    const float scale = 0.5f * (float)(W - 1);

    int   a00[2], a01[2], a10[2], a11[2];
    float w00[2], w01[2], w10[2], w11[2];

    #pragma unroll
    for (int p = 0; p < 2; ++p) {
        // grid = base + normalized flow (align_corners=True convention)
        const float gx = (float)i        + __builtin_fmaf(fx[p], inv, -1.0f);
        const float gy = (float)(j2 + p) + __builtin_fmaf(fy[p], inv, -1.0f);

        float ix = (gx + 1.0f) * scale;
        float iy = (gy + 1.0f) * scale;
        ix = fminf(fmaxf(ix, 0.0f), (float)(W - 1));   // border clamp
        iy = fminf(fmaxf(iy, 0.0f), (float)(H - 1));

        const float ix0f = floorf(ix);
        const float iy0f = floorf(iy);
        const float wx = ix - ix0f;
        const float wy = iy - iy0f;

        const int x0 = (int)ix0f;
        const int y0 = (int)iy0f;
        const int x1 = min(x0 + 1, W - 1);
        const int y1 = min(y0 + 1, H - 1);

        // Bilinear weights computed ONCE per pixel, reused for all 32 channels.
        w00[p] = (1.0f - wx) * (1.0f - wy);
        w01[p] = wx * (1.0f - wy);
        w10[p] = (1.0f - wx) * wy;
        w11[p] = wx * wy;

        // Gather offsets within a channel plane: idx = yy*W + xx
        // (v01 = (iy0, ix1), v10 = (iy1, ix0) per the reference).
        a00[p] = y0 * W + x0;
        a01[p] = y0 * W + x1;
        a10[p] = y1 * W + x0;
        a11[p] = y1 * W + x1;
    }

    const float* __restrict__ pimg = img + (size_t)b * C * HW;
    float* __restrict__ pout = out + (size_t)b * C * HW + (size_t)i * W + j2;

    // 32 channels; unroll 4 -> 32 gathers in flight per thread to hide HBM
    // latency under wave32. Gathers stay regular-temporal (RT): bilinear
    // corners are shared with neighboring pixels -> 192MB L2 reuse.
    #pragma unroll 4
    for (int c = 0; c < C; ++c) {
        const float* __restrict__ pc = pimg + (size_t)c * HW;
        f2 r;
        #pragma unroll
        for (int p = 0; p < 2; ++p) {
            const float v00 = pc[a00[p]];
            const float v01 = pc[a01[p]];
            const float v10 = pc[a10[p]];
            const float v11 = pc[a11[p]];
            float acc = v00 * w00[p];
            acc = __builtin_fmaf(v01, w01[p], acc);
            acc = __builtin_fmaf(v10, w10[p], acc);
            acc = __builtin_fmaf(v11, w11[p], acc);
            r[p] = acc;
        }
        // Streaming write-once -> b64 TH_STORE_NT (don't pollute L2).
        __builtin_nontemporal_store(r, (f2*)(pout + (size_t)c * HW));
    }
}

extern "C" void kernel_launch(void* const* d_in, const int* in_sizes, int n_in,
                              void* d_out, int out_size, void* d_ws, size_t ws_size,
                              hipStream_t stream) {
    const float* img  = (const float*)d_in[0];   // (8,32,512,512) f32
    const float* flow = (const float*)d_in[1];   // (8,2,512,512)  f32
    float* out = (float*)d_out;                  // (8,32,512,512) f32

    const int total = (B * HW) / 2;              // 1,048,576 threads (2 px each)
    const int block = 256;                       // 8 waves (wave32)
    const int grid  = (total + block - 1) / block;
    resample2d_kernel<<<grid, block, 0, stream>>>(img, flow, out);
}